// MultiHeadAttentionBase_65481071396432
// MI455X (gfx1250) — compile-verified
//
#include <hip/hip_runtime.h>

typedef __attribute__((ext_vector_type(2))) float v2f;
typedef __attribute__((ext_vector_type(8))) float v8f;

namespace {
constexpr int Bn  = 4;
constexpr int Hn  = 16;
constexpr int Sn  = 1024;
constexpr int DKn = 64;
constexpr int LDS_STRIDE = 1028;          // 1024 + 4 pad -> rows 8 apart land on different bank halves
constexpr float SCALE  = 0.125f;          // 1/sqrt(64)
constexpr float NEGVAL = -1.0e9f;
}

__global__ __launch_bounds__(256)
void mha_scores_softmax_kernel(const float* __restrict__ Q,
                               const float* __restrict__ K,
                               const int*   __restrict__ mask,
                               float*       __restrict__ out) {
    __shared__ float s_sc[16][LDS_STRIDE];   // one 16-row score stripe, full width

    const int tid  = threadIdx.x;
    const int wave = tid >> 5;
    const int lane = tid & 31;
    const int m    = lane & 15;              // row-within-tile / col-within-tile index
    const int half = lane >> 4;              // selects K sub-pair per WMMA f32 layout

    const int bid   = blockIdx.x;            // 0 .. B*H*(S/16)-1 = 4095
    const int qtile = bid & 63;              // S/16 = 64 q-tiles
    const int bh    = bid >> 6;              // fused (b,h)
    const int bb    = bh >> 4;               // batch index (H=16)
    const int q0    = qtile << 4;

    const float* Qb = Q + ((size_t)bh * Sn + (q0 + m)) * DKn;
    const float* Kb = K + (size_t)bh * Sn * DKn;

    // ---- A fragments: Q[q0+m][4s + 2*half .. +1], s = 0..15 covers DK=64 ----
    v2f a[16];
#pragma unroll
    for (int s = 0; s < 16; ++s)
        a[s] = *(const v2f*)(Qb + 4 * s + 2 * half);

    // ---- phase 1: each wave computes 8 of the 64 column tiles via WMMA ----
    for (int t = 0; t < 8; ++t) {
        const int c0 = (wave + t * 8) << 4;
        const float* Krow = Kb + (size_t)(c0 + m) * DKn;

        v2f bf[16];
#pragma unroll
        for (int s = 0; s < 16; ++s)
            bf[s] = *(const v2f*)(Krow + 4 * s + 2 * half);

        v8f acc = {};
#pragma unroll
        for (int s = 0; s < 16; ++s)
            acc = __builtin_amdgcn_wmma_f32_16x16x4_f32(
                false, a[s], false, bf[s], (short)0, acc, false, false);

        // C/D layout: VGPR v, lanes 0-15 -> row v, lanes 16-31 -> row v+8; col = lane%16
#pragma unroll
        for (int v = 0; v < 8; ++v)
            s_sc[half * 8 + v][c0 + m] = acc[v] * SCALE;
    }
    __syncthreads();

    // ---- phase 2: each wave softmaxes 2 rows (8 waves x 2 = 16 rows) ----
    const size_t plane = (size_t)Bn * Hn * Sn * Sn;
    for (int rr = 0; rr < 2; ++rr) {
        const int row = wave * 2 + rr;
        const int* mrow = mask + ((size_t)bb * Sn + (q0 + row)) * Sn;
        float* srow = s_sc[row];

        // pass A: apply mask (write masked scores back), track row max
        float mx = -3.4e38f;
#pragma unroll
        for (int it = 0; it < 8; ++it) {
            const int c = (it * 32 + lane) * 4;
            float4 x = *(float4*)(srow + c);
            int4  mv = *(const int4*)(mrow + c);
            x.x = mv.x ? x.x : NEGVAL;
            x.y = mv.y ? x.y : NEGVAL;
            x.z = mv.z ? x.z : NEGVAL;
            x.w = mv.w ? x.w : NEGVAL;
            *(float4*)(srow + c) = x;
            mx = fmaxf(mx, fmaxf(fmaxf(x.x, x.y), fmaxf(x.z, x.w)));
        }
#pragma unroll
        for (int off = 16; off >= 1; off >>= 1)
            mx = fmaxf(mx, __shfl_xor(mx, off, 32));

        // pass B: exp-sum
        float sum = 0.0f;
#pragma unroll
        for (int it = 0; it < 8; ++it) {
            const int c = (it * 32 + lane) * 4;
            float4 x = *(float4*)(srow + c);
            sum += expf(x.x - mx) + expf(x.y - mx) + expf(x.z - mx) + expf(x.w - mx);
        }
#pragma unroll
        for (int off = 16; off >= 1; off >>= 1)
            sum += __shfl_xor(sum, off, 32);
        const float inv = 1.0f / sum;

        // pass C: coalesced float4 stores of attn (plane 0) and masked scores (plane 1)
        float* attn = out + ((size_t)bh * Sn + (q0 + row)) * Sn;
        float* sco  = attn + plane;
#pragma unroll
        for (int it = 0; it < 8; ++it) {
            const int c = (it * 32 + lane) * 4;
            float4 x = *(float4*)(srow + c);
            float4 e;
            e.x = expf(x.x - mx) * inv;
            e.y = expf(x.y - mx) * inv;
            e.z = expf(x.z - mx) * inv;
            e.w = expf(x.w - mx) * inv;
            *(float4*)(sco  + c) = x;
            *(float4*)(attn + c) = e;
        }
    }
}

extern "C" void kernel_launch(void* const* d_in, const int* in_sizes, int n_in,
                              void* d_out, int out_size, void* d_ws, size_t ws_size,
                              hipStream_t stream) {
    (void)in_sizes; (void)n_in; (void)out_size; (void)d_ws; (void)ws_size;
    const float* Q    = (const float*)d_in[0];
    const float* K    = (const float*)d_in[1];
    // d_in[2] = value: unused (reference returns (attn, scores) only)
    const int*   mask = (const int*)d_in[3];

    const int grid = Bn * Hn * (Sn / 16);   // 4096 blocks, 256 threads (8 wave32)
    mha_scores_softmax_kernel<<<grid, 256, 0, stream>>>(Q, K, mask, (float*)d_out);
}